// QMFNN_30477087932736
// MI455X (gfx1250) — compile-verified
//
#include <hip/hip_runtime.h>
#include <math.h>

typedef __attribute__((ext_vector_type(2))) float v2f;
typedef __attribute__((ext_vector_type(8))) float v8f;

#define NQ  8
#define DIM 256
#define HALF_PI 1.57079632679489662f

// ---------------- complex helpers ----------------
__device__ __forceinline__ float2 cmul(float2 a, float2 b) {
  return make_float2(a.x * b.x - a.y * b.y, a.x * b.y + a.y * b.x);
}
__device__ __forceinline__ float2 cadd(float2 a, float2 b) {
  return make_float2(a.x + b.x, a.y + b.y);
}

// ---------------- gate primitives on a 256-amplitude statevector in LDS ----
// Wire w <-> bit (7 - w) of the flat index (MSB-first, matches (2,)*8 reshape).
// Pattern per gate: read -> barrier -> write -> barrier (block-wide in-place).

__device__ __forceinline__ void apply1(float2* sv, int tid, int wire,
                                       float2 u00, float2 u01,
                                       float2 u10, float2 u11) {
  int m = 1 << (7 - wire);
  float2 a0 = sv[tid & ~m];
  float2 a1 = sv[tid | m];
  __syncthreads();
  float2 r = (tid & m) ? cadd(cmul(u10, a0), cmul(u11, a1))
                       : cadd(cmul(u00, a0), cmul(u01, a1));
  sv[tid] = r;
  __syncthreads();
}

__device__ __forceinline__ void apply_rx(float2* sv, int tid, int w, float t) {
  float c = cosf(0.5f * t), s = sinf(0.5f * t);
  apply1(sv, tid, w, make_float2(c, 0.f), make_float2(0.f, -s),
                     make_float2(0.f, -s), make_float2(c, 0.f));
}
__device__ __forceinline__ void apply_ry(float2* sv, int tid, int w, float t) {
  float c = cosf(0.5f * t), s = sinf(0.5f * t);
  apply1(sv, tid, w, make_float2(c, 0.f), make_float2(-s, 0.f),
                     make_float2(s, 0.f), make_float2(c, 0.f));
}
__device__ __forceinline__ void apply_rz(float2* sv, int tid, int w, float t) {
  float c = cosf(0.5f * t), s = sinf(0.5f * t);
  // e = exp(-i t/2) = (c, -s); diag(e, conj(e))
  apply1(sv, tid, w, make_float2(c, -s), make_float2(0.f, 0.f),
                     make_float2(0.f, 0.f), make_float2(c, s));
}

// CNOT with control = ctrl wire, target = tgt wire (pure permutation)
__device__ __forceinline__ void apply_cnot(float2* sv, int tid, int ctrl, int tgt) {
  int mc = 1 << (7 - ctrl), mt = 1 << (7 - tgt);
  int src = (tid & mc) ? (tid ^ mt) : tid;
  float2 a = sv[src];
  __syncthreads();
  sv[tid] = a;
  __syncthreads();
}

// exp(-i phi/2 X_i X_j): diagonal c, anti-diagonal (-i s); anti-diagonal
// partner of index s is s with BOTH wire bits flipped.
__device__ __forceinline__ void apply_ising(float2* sv, int tid, int w1, int w2,
                                            float phi) {
  int m = (1 << (7 - w1)) | (1 << (7 - w2));
  float c = cosf(0.5f * phi), s = sinf(0.5f * phi);
  float2 a = sv[tid];
  float2 b = sv[tid ^ m];
  __syncthreads();
  // c*a + (-i*s)*b
  sv[tid] = make_float2(c * a.x + s * b.y, c * a.y - s * b.x);
  __syncthreads();
}

__device__ __forceinline__ void conv_block(float2* sv, int tid, int b1, int b2,
                                           float p0, float p1, float p2) {
  apply_rz(sv, tid, b2, -HALF_PI);
  apply_cnot(sv, tid, b2, b1);
  apply_rz(sv, tid, b1, p0);
  apply_ry(sv, tid, b2, p1);
  apply_cnot(sv, tid, b1, b2);
  apply_ry(sv, tid, b2, p2);
  apply_cnot(sv, tid, b2, b1);
  apply_rz(sv, tid, b1, HALF_PI);
}

__device__ __forceinline__ void pool_block(float2* sv, int tid, int b1, int b2,
                                           float p0, float p1, float p2) {
  apply_rz(sv, tid, b2, -HALF_PI);
  apply_cnot(sv, tid, b2, b1);
  apply_rz(sv, tid, b1, p0);
  apply_ry(sv, tid, b2, p1);
  apply_cnot(sv, tid, b1, b2);
  apply_ry(sv, tid, b2, p2);
}

// ---------------- kernel 1: full statevector simulation --------------------
// p_flip == 0 and af in {0,1}  =>  whole layer is rho = (U psi)(U psi)^dagger.
__global__ __launch_bounds__(DIM) void qsim_state(const float* __restrict__ x,
                                                  const float* __restrict__ wts,
                                                  const float* __restrict__ coup,
                                                  float2* __restrict__ phi_out) {
  __shared__ float2 sv[DIM];
  __shared__ float red[DIM];
  __shared__ float sW[216];
  __shared__ float sC[64];
  __shared__ int sAdj[64];
  int tid = threadIdx.x;

  float xv = x[tid];
  red[tid] = xv * xv;
  if (tid < 210) sW[tid] = wts[tid];
  if (tid < 64) sC[tid] = coup[tid];
  if (tid == 0) {
    float mn = coup[0], mx = coup[0];
    for (int k = 1; k < 64; k++) {
      float v = coup[k];
      mn = fminf(mn, v);
      mx = fmaxf(mx, v);
    }
    for (int k = 0; k < 64; k++) {
      float v = (coup[k] - mn) / (mx - mn);
      sAdj[k] = (v > 0.5f) ? 1 : 0;
    }
  }
  __syncthreads();
  for (int s = 128; s > 0; s >>= 1) {
    if (tid < s) red[tid] += red[tid + s];
    __syncthreads();
  }
  float inv = 1.0f / sqrtf(red[0]);
  sv[tid] = make_float2(xv * inv, 0.f);
  __syncthreads();

  // per-qubit rx, ry, rz (BitFlip channel is identity at p_flip=0)
  for (int q = 0; q < NQ; q++) {
    apply_rx(sv, tid, q, sW[q]);
    apply_ry(sv, tid, q, sW[q + NQ]);
    apply_rz(sv, tid, q, sW[q + 2 * NQ]);
  }
  // ising_xx on all 28 pairs
  for (int i = 0; i < NQ; i++)
    for (int j = i + 1; j < NQ; j++)
      apply_ising(sv, tid, i, j, sC[i * 8 + j]);

  // conditional conv+pool per pair; count-based parameter slicing.
  // sAdj[] is uniform across the block -> barriers inside are safe.
  int count = 0;
  for (int i = 0; i < NQ; i++) {
    for (int j = i + 1; j < NQ; j++) {
      if (sAdj[i * 8 + j]) {
        int idx = 3 * NQ + 6 * count;
        conv_block(sv, tid, i, j, sW[idx], sW[idx + 1], sW[idx + 2]);
        pool_block(sv, tid, i, j, sW[idx + 3], sW[idx + 4], sW[idx + 5]);
        count++;
      }
    }
  }
  phi_out[tid] = sv[tid];
}

// ---------------- kernel 2: rho = phi phi^dagger via V_WMMA_F32_16X16X4_F32 -
// Per 16x16 tile (one wave32):
//   Re = re_i*re_j + im_i*im_j  ->  A=[re,im,0,0](16x4) x B=[re;im;0;0](4x16)
//   Im = im_i*re_j - re_i*im_j  ->  A=[im,-re,0,0]     x same B
// A layout (32-bit 16x4): lanes 0-15 hold K=0,1 in v0,v1; lanes 16-31 K=2,3 (zero).
// D layout: VGPR r -> M=r (lanes 0-15) / M=r+8 (lanes 16-31), N = lane%16.
__global__ __launch_bounds__(32) void outer_wmma(const float2* __restrict__ phi,
                                                 float* __restrict__ out) {
  int tile = blockIdx.x;          // 0..255
  int tm = tile >> 4;             // row tile
  int tn = tile & 15;             // col tile
  int lane = threadIdx.x;         // 0..31

  float aRe = 0.f, aIm = 0.f, bRe = 0.f, bIm = 0.f;
  if (lane < 16) {
    float2 a = phi[tm * 16 + lane];
    aRe = a.x; aIm = a.y;
    float2 b = phi[tn * 16 + lane];
    bRe = b.x; bIm = b.y;
  }

  v2f A_re; A_re[0] = aRe;  A_re[1] = aIm;
  v2f A_im; A_im[0] = aIm;  A_im[1] = -aRe;
  v2f B;    B[0]    = bRe;  B[1]    = bIm;
  v8f zero = {};

  v8f cRe = __builtin_amdgcn_wmma_f32_16x16x4_f32(
      false, A_re, false, B, (short)0, zero, false, false);
  v8f cIm = __builtin_amdgcn_wmma_f32_16x16x4_f32(
      false, A_im, false, B, (short)0, zero, false, false);

  int col = tn * 16 + (lane & 15);
  int rowBase = tm * 16 + ((lane >= 16) ? 8 : 0);
#pragma unroll
  for (int r = 0; r < 8; r++) {
    int o = 2 * ((rowBase + r) * DIM + col);
    out[o]     = cRe[r];
    out[o + 1] = cIm[r];
  }
}

// ---------------- host entry ----------------
extern "C" void kernel_launch(void* const* d_in, const int* in_sizes, int n_in,
                              void* d_out, int out_size, void* d_ws, size_t ws_size,
                              hipStream_t stream) {
  (void)in_sizes; (void)n_in; (void)out_size; (void)ws_size;
  const float* x  = (const float*)d_in[0];   // (256,)
  const float* w  = (const float*)d_in[1];   // (210,)
  const float* cp = (const float*)d_in[2];   // (8,8)
  float2* phi = (float2*)d_ws;               // 256 complex amplitudes

  qsim_state<<<1, DIM, 0, stream>>>(x, w, cp, phi);
  outer_wmma<<<DIM, 32, 0, stream>>>(phi, (float*)d_out);
}